// GNNClassifier_62019327754686
// MI455X (gfx1250) — compile-verified
//
#include <hip/hip_runtime.h>
#include <hip/hip_bf16.h>

typedef float v2f __attribute__((ext_vector_type(2)));
typedef float v8f __attribute__((ext_vector_type(8)));

#define D 128

// ---------------- degree / normalization ----------------

__global__ void k_init_deg(float* __restrict__ deg, int n) {
    int i = blockIdx.x * blockDim.x + threadIdx.x;
    if (i < n) deg[i] = 1.0f;   // self-loop contributes 1
}

__global__ void k_count_deg(const int* __restrict__ col, float* __restrict__ deg, int nE) {
    int e = blockIdx.x * blockDim.x + threadIdx.x;
    if (e < nE) atomicAdd(&deg[col[e]], 1.0f);
}

__global__ void k_dis(const float* __restrict__ deg, float* __restrict__ dis, int n) {
    int i = blockIdx.x * blockDim.x + threadIdx.x;
    if (i < n) dis[i] = rsqrtf(deg[i]);   // deg >= 1 always
}

// ---------------- 128-wide GEMM via V_WMMA_F32_16X16X4_F32 ----------------
// Block = 256 threads = 8 waves. Block computes a 16x128 output stripe:
// wave w owns the 16x16 tile at columns [16w, 16w+16).
// A tile (16x128, optional fused ReLU) and W (128x128) staged in LDS.

__global__ __launch_bounds__(256)
void k_gemm128(const float* __restrict__ A, const float* __restrict__ W,
               float* __restrict__ Y, int M, int relu_in) {
    __shared__ float Ws[D * D];       // 64 KB
    __shared__ float As[16 * 132];    // padded stride kills bank conflicts

    const int t    = threadIdx.x;
    const int wave = t >> 5;
    const int lane = t & 31;
    const int m0   = blockIdx.x * 16;

    // Stage W (row-major K x N), 16 float4 per thread, fully coalesced.
    const float4* W4  = (const float4*)W;
    float4*       Ws4 = (float4*)Ws;
    #pragma unroll
    for (int i = 0; i < 16; ++i) Ws4[t + i * 256] = W4[t + i * 256];

    // Stage A tile: 16 rows x 128 cols = 512 float4, 2 per thread.
    #pragma unroll
    for (int i = 0; i < 2; ++i) {
        int idx4 = t + i * 256;        // 0..511
        int r    = idx4 >> 5;          // tile row 0..15
        int c4   = idx4 & 31;          // float4 column
        int gr   = m0 + r;
        float4 v = make_float4(0.f, 0.f, 0.f, 0.f);
        if (gr < M) v = ((const float4*)A)[(size_t)gr * 32 + c4];
        if (relu_in) {
            v.x = fmaxf(v.x, 0.f); v.y = fmaxf(v.y, 0.f);
            v.z = fmaxf(v.z, 0.f); v.w = fmaxf(v.w, 0.f);
        }
        *(float4*)&As[r * 132 + c4 * 4] = v;
    }
    __syncthreads();

    const int n0 = wave * 16;
    const int mr = lane & 15;   // A row / B-C-D column within tile
    const int kh = lane >> 4;   // which K-pair half this lane holds

    v8f acc = {0.f, 0.f, 0.f, 0.f, 0.f, 0.f, 0.f, 0.f};

    #pragma unroll
    for (int k0 = 0; k0 < D; k0 += 4) {
        int k = k0 + 2 * kh;
        // A fragment: lane (kh,mr) holds A[mr][k], A[mr][k+1]
        v2f a = { As[mr * 132 + k], As[mr * 132 + k + 1] };
        // B fragment: lane (kh,mr) holds W[k][n0+mr], W[k+1][n0+mr]
        v2f b = { Ws[k * D + n0 + mr], Ws[(k + 1) * D + n0 + mr] };
        acc = __builtin_amdgcn_wmma_f32_16x16x4_f32(
            /*neg_a=*/false, a, /*neg_b=*/false, b,
            /*c_mod=*/(short)0, acc, /*reuse_a=*/false, /*reuse_b=*/false);
    }

    // D layout: VGPR r -> row (r + 8*kh), column mr (per ISA 16x16 f32 C/D table)
    const int nn    = n0 + mr;
    const int mbase = m0 + kh * 8;
    #pragma unroll
    for (int r = 0; r < 8; ++r) {
        int gm = mbase + r;
        if (gm < M) Y[(size_t)gm * D + nn] = acc[r];
    }
}

// ---------------- self-loop + bias init: h = dis^2 * xw + b ----------------

__global__ void k_init_h(const float4* __restrict__ xw, const float* __restrict__ dis,
                         const float* __restrict__ bias, float4* __restrict__ h, int n) {
    int idx = blockIdx.x * blockDim.x + threadIdx.x;   // over n*32 float4s
    if (idx >= n * 32) return;
    int i  = idx >> 5;
    int d4 = idx & 31;
    float s = dis[i]; s = s * s;
    float4 v = xw[idx];
    float4 b = ((const float4*)bias)[d4];
    h[idx] = make_float4(s * v.x + b.x, s * v.y + b.y, s * v.z + b.z, s * v.w + b.w);
}

// ---------------- edge scatter: out[col] += dis[row]*dis[col] * xw[row] ----------------
// One wave per edge, one float4 per lane (128 floats = full feature row).

__global__ __launch_bounds__(256)
void k_scatter(const float4* __restrict__ xw, const int* __restrict__ row,
               const int* __restrict__ col, const float* __restrict__ dis,
               float* __restrict__ out, int nE) {
    int e = blockIdx.x * 8 + (threadIdx.x >> 5);
    if (e >= nE) return;
    int lane = threadIdx.x & 31;
    int r = row[e];
    int c = col[e];
    float norm = dis[r] * dis[c];
    float4 v = xw[(size_t)r * 32 + lane];
    float* dst = out + (size_t)c * D + lane * 4;
    atomicAdd(dst + 0, norm * v.x);
    atomicAdd(dst + 1, norm * v.y);
    atomicAdd(dst + 2, norm * v.z);
    atomicAdd(dst + 3, norm * v.w);
}

// ---------------- pooling ----------------

__global__ void k_zero(float* __restrict__ p, int n) {
    int i = blockIdx.x * blockDim.x + threadIdx.x;
    if (i < n) p[i] = 0.0f;
}

__global__ __launch_bounds__(256)
void k_pool(const float4* __restrict__ h, const int* __restrict__ batch,
            float* __restrict__ sums, float* __restrict__ cnt, int n) {
    int i = blockIdx.x * 8 + (threadIdx.x >> 5);
    if (i >= n) return;
    int lane = threadIdx.x & 31;
    int g = batch[i];
    float4 v = h[(size_t)i * 32 + lane];
    float* dst = sums + (size_t)g * D + lane * 4;
    atomicAdd(dst + 0, fmaxf(v.x, 0.f));   // fused ReLU
    atomicAdd(dst + 1, fmaxf(v.y, 0.f));
    atomicAdd(dst + 2, fmaxf(v.z, 0.f));
    atomicAdd(dst + 3, fmaxf(v.w, 0.f));
    if (lane == 0) atomicAdd(&cnt[g], 1.0f);
}

// ---------------- head: mean, FC, log_softmax ----------------

__global__ void k_head(const float* __restrict__ sums, const float* __restrict__ cnt,
                       const float* __restrict__ Wfc, const float* __restrict__ bfc,
                       float* __restrict__ out, int G) {
    int g = threadIdx.x;
    if (g >= G) return;
    float inv = 1.0f / fmaxf(cnt[g], 1.0f);
    float l[10];
    #pragma unroll
    for (int j = 0; j < 10; ++j) l[j] = bfc[j];
    for (int d = 0; d < D; ++d) {
        float e = sums[g * D + d] * inv;
        #pragma unroll
        for (int j = 0; j < 10; ++j) l[j] += e * Wfc[d * 10 + j];
    }
    float m = l[0];
    #pragma unroll
    for (int j = 1; j < 10; ++j) m = fmaxf(m, l[j]);
    float s = 0.f;
    #pragma unroll
    for (int j = 0; j < 10; ++j) s += __expf(l[j] - m);
    float lse = __logf(s);
    #pragma unroll
    for (int j = 0; j < 10; ++j) out[g * 10 + j] = l[j] - m - lse;
}

// ---------------- launch ----------------

extern "C" void kernel_launch(void* const* d_in, const int* in_sizes, int n_in,
                              void* d_out, int out_size, void* d_ws, size_t ws_size,
                              hipStream_t stream) {
    const float* x     = (const float*)d_in[0];
    const int*   ei    = (const int*)d_in[1];
    const int*   batch = (const int*)d_in[2];
    const float* W1    = (const float*)d_in[3];
    const float* b1    = (const float*)d_in[4];
    const float* W2    = (const float*)d_in[5];
    const float* b2    = (const float*)d_in[6];
    const float* Wfc   = (const float*)d_in[7];
    const float* bfc   = (const float*)d_in[8];
    float* out = (float*)d_out;

    const int N = in_sizes[2];       // 50000 nodes (batch vector length)
    const int E = in_sizes[1] / 2;   // 800000 edges
    const int G = 64;
    const int* row = ei;             // edge_index[0]
    const int* col = ei + E;         // edge_index[1]

    float* ws   = (float*)d_ws;
    float* deg  = ws;                    // N
    float* dis  = ws + N;                // N
    float* sums = ws + 2 * (size_t)N;    // G*D
    float* cnt  = sums + G * D;          // G
    size_t off  = (size_t)(2 * (size_t)N + G * D + G);
    off = (off + 255) & ~(size_t)255;    // keep float4 alignment
    float* bufA = ws + off;              // N*D  (xw / xw2)
    float* bufB = bufA + (size_t)N * D;  // N*D  (h / h2)

    const int nb = (N + 255) / 256;
    const int mb = (N + 15) / 16;            // GEMM row-stripes
    const int hb = (N * 32 + 255) / 256;     // float4 elementwise
    const int eb = (E + 7) / 8;              // wave-per-edge
    const int pb = (N + 7) / 8;              // wave-per-node

    // normalization
    k_init_deg <<<nb, 256, 0, stream>>>(deg, N);
    k_count_deg<<<(E + 255) / 256, 256, 0, stream>>>(col, deg, E);
    k_dis      <<<nb, 256, 0, stream>>>(deg, dis, N);

    // layer 1: xw = x @ W1 ; h = dis^2*xw + b1 ; scatter edges
    k_gemm128<<<mb, 256, 0, stream>>>(x, W1, bufA, N, 0);
    k_init_h <<<hb, 256, 0, stream>>>((const float4*)bufA, dis, b1, (float4*)bufB, N);
    k_scatter<<<eb, 256, 0, stream>>>((const float4*)bufA, row, col, dis, bufB, E);

    // layer 2: xw2 = relu(h) @ W2 ; h2 = dis^2*xw2 + b2 ; scatter edges
    k_gemm128<<<mb, 256, 0, stream>>>(bufB, W2, bufA, N, 1);
    k_init_h <<<hb, 256, 0, stream>>>((const float4*)bufA, dis, b2, (float4*)bufB, N);
    k_scatter<<<eb, 256, 0, stream>>>((const float4*)bufA, row, col, dis, bufB, E);

    // mean pool (ReLU fused) + head
    k_zero<<<(G * D + G + 255) / 256, 256, 0, stream>>>(sums, G * D + G);
    k_pool<<<pb, 256, 0, stream>>>((const float4*)bufB, batch, sums, cnt, N);
    k_head<<<1, 64, 0, stream>>>(sums, cnt, Wfc, bfc, out, G);
}